// MARA_45921790328877
// MI455X (gfx1250) — compile-verified
//
#include <hip/hip_runtime.h>
#include <math.h>

typedef __attribute__((ext_vector_type(2))) float v2f;
typedef __attribute__((ext_vector_type(8))) float v8f;

#define BM 64
#define BN 64
#define BK 16
#define ASTRIDE 20   // LDS row stride for A (floats), multiple of 4 for float4 stores
#define BPS 160      // LDS stride per K-pair for B (floats); 160 % 64 == 32 -> halves on disjoint banks

// ---------- graph-norm kernels ----------
__global__ void k_fill(float* __restrict__ p, float v, int n) {
    int i = blockIdx.x * blockDim.x + threadIdx.x;
    if (i < n) p[i] = v;
}

__global__ void k_deg(float* __restrict__ deg, const int* __restrict__ edges, int nE) {
    int e = blockIdx.x * blockDim.x + threadIdx.x;
    if (e < nE) atomicAdd(&deg[edges[2 * e + 1]], 1.0f);
}

__global__ void k_dinv(float* __restrict__ deg, int n) {
    int i = blockIdx.x * blockDim.x + threadIdx.x;
    if (i < n) {
        float d = deg[i];
        deg[i] = (d > 0.0f) ? rsqrtf(d) : 0.0f;
    }
}

// ---------- fp32 WMMA GEMM: C[M,N] = A[M,K] @ B[K,N] ----------
// Block: 256 threads = 8 waves. Wave grid 4(M) x 2(N); wave tile 16x32 (two 16x16 wmma tiles).
// Requires K % 16 == 0 and N % 4 == 0 (true for all layers: K=256/512/256, N=512/256/52).
// Out-of-range tile rows/cols are index-CLAMPED, not zeroed: they only produce
// C entries that the guarded store phase never writes, so no divergent loads.
__global__ __launch_bounds__(256) void k_gemm_wmma(const float* __restrict__ A,
                                                   const float* __restrict__ B,
                                                   float* __restrict__ C,
                                                   int M, int N, int K) {
    __shared__ float As[BM * ASTRIDE];
    __shared__ float BsP[8 * BPS];  // 8 K-pairs; (k,n) at [(k>>1)*BPS + 2n + (k&1)]

    const int tid   = threadIdx.x;
    const int lane  = tid & 31;
    const int w     = tid >> 5;
    const int waveM = w & 3;   // 0..3
    const int waveN = w >> 2;  // 0..1
    const int mBase = blockIdx.y * BM;
    const int nBase = blockIdx.x * BN;

    // Cooperative load assignments (clamped -> unconditional b128 loads)
    const int aRow = tid >> 2;          // 0..63
    const int aK   = (tid & 3) << 2;    // 0,4,8,12
    const int gmC  = min(mBase + aRow, M - 1);
    const int bK   = tid >> 4;          // 0..15
    const int bN   = (tid & 15) << 2;   // 0..60
    const int gnC  = min(nBase + bN, N - 4);   // N%4==0 keeps 16B alignment
    const int bPair = bK >> 1;
    const int bSel  = bK & 1;

    // Fragment indices (per the 16x4 A / 4x16 B / 16x16 C VGPR layouts)
    const int half = lane >> 4;         // 0: K pair {0,1}; 1: K pair {2,3}
    const int l15  = lane & 15;
    const int mrow = waveM * 16 + l15;  // A row within block tile
    const int n0   = waveN * 32 + l15;  // B/C col within block tile

    v8f acc0 = {};
    v8f acc1 = {};

    for (int kt = 0; kt < K; kt += BK) {
        __syncthreads();  // protect LDS from previous iteration's readers

        // A tile: BM x BK (K%4==0, row strides 256/512 floats -> 16B aligned)
        float4 av = *(const float4*)(A + (size_t)gmC * K + kt + aK);
        *(float4*)(&As[aRow * ASTRIDE + aK]) = av;

        // B tile: BK x BN, K-pair-interleaved so fragments are contiguous b64 reads
        float4 bv = *(const float4*)(B + (size_t)(kt + bK) * N + gnC);
        {
            float* bp = &BsP[bPair * BPS + (bN << 1) + bSel];
            bp[0] = bv.x; bp[2] = bv.y; bp[4] = bv.z; bp[6] = bv.w;
        }
        __syncthreads();

#pragma unroll
        for (int kk = 0; kk < 4; ++kk) {
            int ka   = kk * 4 + half * 2;      // even
            int pair = (kk << 1) + half;       // ka >> 1
            v2f a  = *(const v2f*)(&As[mrow * ASTRIDE + ka]);       // a.x=K=ka, a.y=K=ka+1
            v2f b0 = *(const v2f*)(&BsP[pair * BPS + (n0 << 1)]);   // (ka,n0),(ka+1,n0)
            v2f b1 = *(const v2f*)(&BsP[pair * BPS + (n0 << 1) + 32]);
            acc0 = __builtin_amdgcn_wmma_f32_16x16x4_f32(false, a, false, b0,
                                                         (short)0, acc0, false, false);
            acc1 = __builtin_amdgcn_wmma_f32_16x16x4_f32(false, a, false, b1,
                                                         (short)0, acc1, false, false);
        }
    }

    // C VGPR layout: VGPR i -> row (i or i+8 for lanes>=16), col = lane&15 (+16 for acc1)
    const int n0g   = nBase + waveN * 32 + l15;
    const int mrow0 = mBase + waveM * 16 + half * 8;
#pragma unroll
    for (int i = 0; i < 8; ++i) {
        int m = mrow0 + i;
        if (m < M) {
            if (n0g < N)      C[(size_t)m * N + n0g]      = acc0[i];
            if (n0g + 16 < N) C[(size_t)m * N + n0g + 16] = acc1[i];
        }
    }
}

// ---------- aggregation ----------
// out[i,:] = xw[i,:] * dinv[i]^2  (self-loop contribution); i = blockIdx.y
__global__ void k_agg_init(float* __restrict__ out, const float* __restrict__ xw,
                           const float* __restrict__ dinv, int D) {
    int i = blockIdx.y;
    int f = blockIdx.x * blockDim.x + threadIdx.x;
    if (f >= D) return;
    float di = dinv[i];
    size_t idx = (size_t)i * D + f;
    out[idx] = xw[idx] * di * di;
}

// one block per edge: out[dst,:] += xw[src,:] * dinv[src]*dinv[dst]
__global__ void k_agg_edges(float* __restrict__ out, const float* __restrict__ xw,
                            const float* __restrict__ dinv,
                            const int* __restrict__ edges, int nE, int D) {
    int e = blockIdx.x;
    if (e >= nE) return;
    int s = edges[2 * e];
    int d = edges[2 * e + 1];
    float nrm = dinv[s] * dinv[d];
    const float* xs = xw + (size_t)s * D;
    float* od = out + (size_t)d * D;
    for (int f = threadIdx.x; f < D; f += blockDim.x)
        atomicAdd(&od[f], xs[f] * nrm);
}

// h = clip(h + b, 0, 6); i = blockIdx.y
__global__ void k_bias_act(float* __restrict__ h, const float* __restrict__ b, int D) {
    int i = blockIdx.y;
    int f = blockIdx.x * blockDim.x + threadIdx.x;
    if (f >= D) return;
    size_t idx = (size_t)i * D + f;
    float v = h[idx] + b[f];
    h[idx] = fminf(fmaxf(v, 0.0f), 6.0f);
}

// out = sigmoid(h @ wc + bc), tiny (D3=52, NC=3) per-node kernel
__global__ void k_cls(const float* __restrict__ h, const float* __restrict__ wc,
                      const float* __restrict__ bc, float* __restrict__ out,
                      int n, int D3, int NC) {
    int i = blockIdx.x * blockDim.x + threadIdx.x;
    if (i >= n) return;
    float acc[8];
    for (int c = 0; c < NC; ++c) acc[c] = bc[c];
    const float* hi = h + (size_t)i * D3;
    for (int f = 0; f < D3; ++f) {
        float hv = hi[f];
        for (int c = 0; c < NC; ++c) acc[c] += hv * wc[f * NC + c];
    }
    for (int c = 0; c < NC; ++c)
        out[(size_t)i * NC + c] = 1.0f / (1.0f + expf(-acc[c]));
}

extern "C" void kernel_launch(void* const* d_in, const int* in_sizes, int n_in,
                              void* d_out, int out_size, void* d_ws, size_t ws_size,
                              hipStream_t stream) {
    const float* x     = (const float*)d_in[0];
    const int*   intra = (const int*)d_in[1];
    const int*   cross = (const int*)d_in[2];
    const float* w1 = (const float*)d_in[3];
    const float* b1 = (const float*)d_in[4];
    const float* w2 = (const float*)d_in[5];
    const float* b2 = (const float*)d_in[6];
    const float* w3 = (const float*)d_in[7];
    const float* b3 = (const float*)d_in[8];
    const float* wc = (const float*)d_in[9];
    const float* bc = (const float*)d_in[10];
    float* out = (float*)d_out;

    const int D1  = in_sizes[4];           // 512
    const int D2  = in_sizes[6];           // 256
    const int D3  = in_sizes[8];           // 52
    const int NC  = in_sizes[10];          // 3
    const int DIN = in_sizes[3] / D1;      // 256
    const int n   = in_sizes[0] / DIN;     // 50000
    const int e1  = in_sizes[1] / 2;       // 400000
    const int e2  = in_sizes[2] / 2;       // 400000

    // workspace layout: dinv | bufP (xw) | bufQ (agg/h), ping-pong
    float* ws = (float*)d_ws;
    size_t dinvSz = ((size_t)n + 255) & ~(size_t)255;
    size_t bufSz  = (size_t)n * (size_t)(D1 > DIN ? D1 : DIN);
    float* dinv = ws;
    float* bufP = ws + dinvSz;
    float* bufQ = bufP + bufSz;

    const int T = 256;

    // degrees (self-loop -> init 1.0) and symmetric norm
    k_fill<<<(n + T - 1) / T, T, 0, stream>>>(dinv, 1.0f, n);
    k_deg<<<(e1 + T - 1) / T, T, 0, stream>>>(dinv, intra, e1);
    k_deg<<<(e2 + T - 1) / T, T, 0, stream>>>(dinv, cross, e2);
    k_dinv<<<(n + T - 1) / T, T, 0, stream>>>(dinv, n);

    struct Layer { const float* h; int K; const float* W; const float* b; int D; };
    Layer layers[3] = {
        { x,    DIN, w1, b1, D1 },
        { bufQ, D1,  w2, b2, D2 },
        { bufQ, D2,  w3, b3, D3 },
    };
    for (int l = 0; l < 3; ++l) {
        const Layer& L = layers[l];
        dim3 grid((L.D + BN - 1) / BN, (n + BM - 1) / BM);
        k_gemm_wmma<<<grid, 256, 0, stream>>>(L.h, L.W, bufP, n, L.D, L.K);
        // elementwise kernels: feature -> x, node -> y (no runtime div/mod)
        int fbs = L.D >= 256 ? 256 : ((L.D + 31) & ~31);
        dim3 egrid((L.D + fbs - 1) / fbs, n);
        // edge-scatter block size: cover D with 32-lane granularity, max 128
        int ebs = L.D >= 128 ? 128 : ((L.D + 31) & ~31);
        k_agg_init<<<egrid, fbs, 0, stream>>>(bufQ, bufP, dinv, L.D);
        k_agg_edges<<<e1, ebs, 0, stream>>>(bufQ, bufP, dinv, intra, e1, L.D);
        k_agg_edges<<<e2, ebs, 0, stream>>>(bufQ, bufP, dinv, cross, e2, L.D);
        k_bias_act<<<egrid, fbs, 0, stream>>>(bufQ, L.b, L.D);
    }

    // classifier + sigmoid
    k_cls<<<(n + T - 1) / T, T, 0, stream>>>(bufQ, wc, bc, out, n, D3, NC);

    // pass-through outputs: edge arrays, bitwise into d_out after the probabilities
    char* oc = (char*)d_out;
    size_t off = (size_t)n * (size_t)NC * sizeof(float);
    hipMemcpyAsync(oc + off, d_in[1], (size_t)in_sizes[1] * sizeof(int),
                   hipMemcpyDeviceToDevice, stream);
    off += (size_t)in_sizes[1] * sizeof(int);
    hipMemcpyAsync(oc + off, d_in[2], (size_t)in_sizes[2] * sizeof(int),
                   hipMemcpyDeviceToDevice, stream);
}